// Qwen3NextGatedDeltaNetPrefill_62337155334414
// MI455X (gfx1250) — compile-verified
//
#include <hip/hip_runtime.h>
#include <hip/hip_bf16.h>

// Qwen3-Next GatedDeltaNet prefill for gfx1250 (MI455X), chunked delta rule.
//
// grid = 32 heads x 4 DV-splits = 128 blocks, 256 threads (8 wave32) each.
// Per block: state slice S [128 x 32] f32 lives in LDS across the sequential
// 64-chunk scan. All chunk math is f32 WMMA (v_wmma_f32_16x16x4_f32):
//   N = -beta_i e^{gc_i-gc_j} (k_i.k_j)  (strict lower)       [K K^T matmul]
//   T = (I-N)^{-1} = prod_k (I + N^{2^k})  (5 squarings)      [64x64 matmuls]
//   rhs = beta (V - e^{gc} K S0)                               [K S matmul]
//   U = T rhs ;  O = scale (e^{gc} Q S0 + mask(QK^T) U)        [matmuls]
//   S = e^{gcL} S0 + Ktilde^T U                                [matmul]
// Conv1d(+SiLU) and l2norm fused into the chunk tile loader -> no workspace.
// Two-tile stages run both accumulator chains unconditionally in one k-loop
// (shared B frags, causal mask applied at the store transform) so the WMMA
// stream stays branch-free and software-pipelined; next chunk's global slab
// is prefetched (global_prefetch_b8) during compute.

typedef float v2f __attribute__((ext_vector_type(2)));
typedef float v8f __attribute__((ext_vector_type(8)));

#define T_LEN   4096
#define HV_N    32
#define QKV_W   8192
#define CH_L    64
#define N_CHUNK (T_LEN / CH_L)
#define NSPLIT  4
#define WV      32      // DV columns handled per block
#define LDQ     132     // LDS stride (floats) for 128-wide tiles
#define LDT     68      // LDS stride for 64-wide tiles
#define LDV     36      // LDS stride for 32-wide tiles

#define WMMA_F32(a, b, c) \
  __builtin_amdgcn_wmma_f32_16x16x4_f32(false, (a), false, (b), (short)0, (c), false, false)

__device__ __forceinline__ float sigm_(float x) { return 1.f / (1.f + expf(-x)); }

// A fragment: 16x4 tile at (row0,k0). lanes 0-15: K=k0,k0+1; lanes 16-31: K=k0+2,k0+3
__device__ __forceinline__ v2f ldA(const float* P, int ld, int row0, int k0, int lh, int hi) {
  int r = row0 + lh, k = k0 + 2 * hi;
  v2f a; a.x = P[r * ld + k]; a.y = P[r * ld + k + 1]; return a;
}
// B fragment: 4x16 tile at (k0,col0), row-major source
__device__ __forceinline__ v2f ldB(const float* P, int ld, int k0, int col0, int lh, int hi) {
  int k = k0 + 2 * hi, c = col0 + lh;
  v2f b; b.x = P[k * ld + c]; b.y = P[(k + 1) * ld + c]; return b;
}
// B fragment where B = P^T (B[k][c] = P[c][k])
__device__ __forceinline__ v2f ldBT(const float* P, int ld, int k0, int col0, int lh, int hi) {
  int k = k0 + 2 * hi, c = col0 + lh;
  v2f b; b.x = P[c * ld + k]; b.y = P[c * ld + k + 1]; return b;
}
// C/D store: VGPR r -> row row0+r+8*hi, col col0+lane&15
__device__ __forceinline__ void stC(float* P, int ld, int row0, int col0, int lh, int hi, v8f c) {
  int col = col0 + lh;
#pragma unroll
  for (int r = 0; r < 8; ++r) P[(row0 + r + 8 * hi) * ld + col] = c[r];
}

__global__ __launch_bounds__(256) void gdn_prefill_kernel(
    const float* __restrict__ mixed, const float* __restrict__ bgate,
    const float* __restrict__ again, const float* __restrict__ convw,
    const float* __restrict__ dtb, const float* __restrict__ alogv,
    float* __restrict__ out) {
  __shared__ float S[128 * LDV];           // state slice [128 x WV]
  __shared__ float Kt[CH_L * LDQ];         // keys   [64 x 128] (l2-normed)
  __shared__ float Qt[CH_L * LDQ];         // queries[64 x 128] (l2-normed)
  __shared__ float Vt[CH_L * LDV];         // values -> rhs in place [64 x WV]
  __shared__ float Ut[CH_L * LDV];         // pseudo-values U [64 x WV]
  __shared__ float NN[CH_L * LDT];         // N matrix / ping / attn matrix
  __shared__ float MM[CH_L * LDT];         // pong
  __shared__ float TAb[CH_L * LDT];        // T ping
  __shared__ float TBb[CH_L * LDT];        // T pong
  __shared__ float gcv[CH_L], betav[CH_L], egc[CH_L], escl[CH_L];
  __shared__ float rnq[CH_L], rnk[CH_L];

  const int head = (int)blockIdx.x >> 2;
  const int split = (int)blockIdx.x & 3;
  const int kh = head >> 1;                // GQA: 2 value heads per key head
  const int tid = threadIdx.x;
  const int wid = tid >> 5, lane = tid & 31;
  const int lh = lane & 15, hi = lane >> 4;

  const int qcol = kh * 128;
  const int kcol = 2048 + kh * 128;
  const int vcol = 4096 + head * 128 + split * WV;
  const float Aexp = expf(alogv[head]);
  const float dbias = dtb[head];
  const float scaleO = 0.08838834764831845f;  // 128^-0.5

  for (int i = tid; i < 128 * LDV; i += 256) S[i] = 0.f;

  for (int c = 0; c < N_CHUNK; ++c) {
    __syncthreads();
    const int tbase = c * CH_L;

    // ---- fused conv1d + SiLU loads for q, k (full 128 cols) ----
    for (int e = tid; e < CH_L * 128; e += 256) {
      int i = e >> 7, d = e & 127;
      int t = tbase + i;
      float aq = 0.f, ak = 0.f;
#pragma unroll
      for (int j = 0; j < 4; ++j) {
        int tt = t - 3 + j;
        if (tt >= 0) {
          aq += mixed[(size_t)tt * QKV_W + qcol + d] * convw[(qcol + d) * 4 + j];
          ak += mixed[(size_t)tt * QKV_W + kcol + d] * convw[(kcol + d) * 4 + j];
        }
      }
      Qt[i * LDQ + d] = aq * sigm_(aq);
      Kt[i * LDQ + d] = ak * sigm_(ak);
    }
    // ---- v slice (WV cols) ----
    for (int e = tid; e < CH_L * WV; e += 256) {
      int i = e / WV, d = e % WV;
      int t = tbase + i;
      float av = 0.f;
#pragma unroll
      for (int j = 0; j < 4; ++j) {
        int tt = t - 3 + j;
        if (tt >= 0) av += mixed[(size_t)tt * QKV_W + vcol + d] * convw[(vcol + d) * 4 + j];
      }
      Vt[i * LDV + d] = av * sigm_(av);
    }
    // ---- gating: g = -exp(Alog)*softplus(a+dt_bias), beta = sigmoid(b) ----
    if (tid < CH_L) {
      int t = tbase + tid;
      float av = again[(size_t)t * HV_N + head] + dbias;
      float sp = (av > 20.f) ? av : log1pf(expf(av));
      gcv[tid] = -Aexp * sp;  // per-step g; prefix-summed below
      betav[tid] = sigm_(bgate[(size_t)t * HV_N + head]);
    }
    // ---- prefetch next chunk's q/k/v slab into cache (global_prefetch_b8) --
    if (c + 1 < N_CHUNK) {
      int nb = tbase + CH_L;
      for (int e = tid; e < 576; e += 256) {
        size_t addr;
        if (e < 256) {            // q rows: 4 x 128B lines per row
          addr = (size_t)(nb + (e >> 2)) * QKV_W + qcol + (e & 3) * 32;
        } else if (e < 512) {     // k rows
          int f = e - 256;
          addr = (size_t)(nb + (f >> 2)) * QKV_W + kcol + (f & 3) * 32;
        } else {                  // v rows: 1 line per row
          addr = (size_t)(nb + (e - 512)) * QKV_W + vcol;
        }
        __builtin_prefetch(&mixed[addr], 0, 3);
      }
    }
    __syncthreads();
    // ---- l2 norms of q/k rows; inclusive prefix sum of g ----
    if (tid < 128) {
      int r = tid & 63;
      const float* P = (tid < 64) ? Qt : Kt;
      float s = 0.f;
      for (int d = 0; d < 128; ++d) { float x = P[r * LDQ + d]; s += x * x; }
      float rn = rsqrtf(s + 1e-6f);
      if (tid < 64) rnq[r] = rn; else rnk[r] = rn;
    }
    if (tid == 0) {
      float s = 0.f;
      for (int i = 0; i < CH_L; ++i) { s += gcv[i]; gcv[i] = s; }
    }
    __syncthreads();
    for (int e = tid; e < CH_L * 128; e += 256) {
      int i = e >> 7, d = e & 127;
      Qt[i * LDQ + d] *= rnq[i];
      Kt[i * LDQ + d] *= rnk[i];
    }
    if (tid < CH_L) {
      egc[tid] = expf(gcv[tid]);                  // e^{gc_i} (<= 1)
      escl[tid] = expf(gcv[CH_L - 1] - gcv[tid]); // e^{gcL - gc_i} (<= 1)
    }
    __syncthreads();

    // Wave's two 16x16 tiles share a tile column -> shared B frags, 2 chains.
    const int tj4 = wid & 3, ti0 = wid >> 2, ti1 = ti0 + 2;

    // ---- N = -beta_i e^{gc_i-gc_j} (K K^T), strictly lower ----
    // Both chains computed unconditionally; causal mask applied at the store.
    {
      v8f acc0 = {}, acc1 = {};
      for (int kk = 0; kk < 128; kk += 4) {
        v2f b = ldBT(Kt, LDQ, kk, tj4 * 16, lh, hi);
        v2f a0 = ldA(Kt, LDQ, ti0 * 16, kk, lh, hi);
        v2f a1 = ldA(Kt, LDQ, ti1 * 16, kk, lh, hi);
        acc0 = WMMA_F32(a0, b, acc0);
        acc1 = WMMA_F32(a1, b, acc1);
      }
      int jg = tj4 * 16 + lh;
#pragma unroll
      for (int r = 0; r < 8; ++r) {
        int i0 = ti0 * 16 + r + 8 * hi, i1 = ti1 * 16 + r + 8 * hi;
        NN[i0 * LDT + jg] = (jg < i0) ? (-betav[i0] * expf(gcv[i0] - gcv[jg]) * acc0[r]) : 0.f;
        NN[i1 * LDT + jg] = (jg < i1) ? (-betav[i1] * expf(gcv[i1] - gcv[jg]) * acc1[r]) : 0.f;
      }
    }
    __syncthreads();
    // ---- T = I + N ----
    for (int e = tid; e < CH_L * CH_L; e += 256) {
      int i = e >> 6, j = e & 63;
      TAb[i * LDT + j] = NN[i * LDT + j] + ((i == j) ? 1.f : 0.f);
    }
    __syncthreads();
    // ---- T = prod (I + N^{2^k}) via 5 squarings (log-depth inverse) ----
    float *Mc = NN, *Mn = MM, *Tc = TAb, *Tn = TBb;
    for (int it = 0; it < 5; ++it) {
      {  // Mn = Mc @ Mc (two tiles, shared B)
        v8f acc0 = {}, acc1 = {};
        for (int kk = 0; kk < 64; kk += 4) {
          v2f b = ldB(Mc, LDT, kk, tj4 * 16, lh, hi);
          v2f a0 = ldA(Mc, LDT, ti0 * 16, kk, lh, hi);
          v2f a1 = ldA(Mc, LDT, ti1 * 16, kk, lh, hi);
          acc0 = WMMA_F32(a0, b, acc0);
          acc1 = WMMA_F32(a1, b, acc1);
        }
        stC(Mn, LDT, ti0 * 16, tj4 * 16, lh, hi, acc0);
        stC(Mn, LDT, ti1 * 16, tj4 * 16, lh, hi, acc1);
      }
      __syncthreads();
      {  // Tn = Tc + Tc @ Mn (two tiles, shared B)
        v8f acc0, acc1;
        int jg = tj4 * 16 + lh;
#pragma unroll
        for (int r = 0; r < 8; ++r) {
          acc0[r] = Tc[(ti0 * 16 + r + 8 * hi) * LDT + jg];
          acc1[r] = Tc[(ti1 * 16 + r + 8 * hi) * LDT + jg];
        }
        for (int kk = 0; kk < 64; kk += 4) {
          v2f b = ldB(Mn, LDT, kk, tj4 * 16, lh, hi);
          v2f a0 = ldA(Tc, LDT, ti0 * 16, kk, lh, hi);
          v2f a1 = ldA(Tc, LDT, ti1 * 16, kk, lh, hi);
          acc0 = WMMA_F32(a0, b, acc0);
          acc1 = WMMA_F32(a1, b, acc1);
        }
        stC(Tn, LDT, ti0 * 16, tj4 * 16, lh, hi, acc0);
        stC(Tn, LDT, ti1 * 16, tj4 * 16, lh, hi, acc1);
      }
      __syncthreads();
      float* t0 = Mc; Mc = Mn; Mn = t0;
      float* t1 = Tc; Tc = Tn; Tn = t1;
    }
    float* ATT = Mn;  // free 64x64 buffer

    // ---- attn = mask_{j<=i}( e^{gc_i-gc_j} (Q K^T) ) ----
    {
      v8f acc0 = {}, acc1 = {};
      for (int kk = 0; kk < 128; kk += 4) {
        v2f b = ldBT(Kt, LDQ, kk, tj4 * 16, lh, hi);
        v2f a0 = ldA(Qt, LDQ, ti0 * 16, kk, lh, hi);
        v2f a1 = ldA(Qt, LDQ, ti1 * 16, kk, lh, hi);
        acc0 = WMMA_F32(a0, b, acc0);
        acc1 = WMMA_F32(a1, b, acc1);
      }
      int jg = tj4 * 16 + lh;
#pragma unroll
      for (int r = 0; r < 8; ++r) {
        int i0 = ti0 * 16 + r + 8 * hi, i1 = ti1 * 16 + r + 8 * hi;
        ATT[i0 * LDT + jg] = (jg <= i0) ? (expf(gcv[i0] - gcv[jg]) * acc0[r]) : 0.f;
        ATT[i1 * LDT + jg] = (jg <= i1) ? (expf(gcv[i1] - gcv[jg]) * acc1[r]) : 0.f;
      }
    }
    // ---- rhs = beta (V - e^{gc} K S0), written into Vt in place ----
    {
      int ti = wid >> 1, tj = wid & 1;
      v8f acc = {};
      for (int kk = 0; kk < 128; kk += 4) {
        v2f a = ldA(Kt, LDQ, ti * 16, kk, lh, hi);
        v2f b = ldB(S, LDV, kk, tj * 16, lh, hi);
        acc = WMMA_F32(a, b, acc);
      }
      int jl = tj * 16 + lh;
#pragma unroll
      for (int r = 0; r < 8; ++r) {
        int il = ti * 16 + r + 8 * hi;
        Vt[il * LDV + jl] = betav[il] * (Vt[il * LDV + jl] - egc[il] * acc[r]);
      }
    }
    __syncthreads();
    // ---- U = T @ rhs ----
    {
      int ti = wid >> 1, tj = wid & 1;
      v8f acc = {};
      for (int kk = 0; kk < 64; kk += 4) {
        v2f a = ldA(Tc, LDT, ti * 16, kk, lh, hi);
        v2f b = ldB(Vt, LDV, kk, tj * 16, lh, hi);
        acc = WMMA_F32(a, b, acc);
      }
      stC(Ut, LDV, ti * 16, tj * 16, lh, hi, acc);
    }
    __syncthreads();
    // ---- O = scale * (e^{gc} (Q S0) + attn @ U) -> global ----
    {
      int ti = wid >> 1, tj = wid & 1;
      v8f acc = {};
      for (int kk = 0; kk < 128; kk += 4) {
        v2f a = ldA(Qt, LDQ, ti * 16, kk, lh, hi);
        v2f b = ldB(S, LDV, kk, tj * 16, lh, hi);
        acc = WMMA_F32(a, b, acc);
      }
#pragma unroll
      for (int r = 0; r < 8; ++r) acc[r] *= egc[ti * 16 + r + 8 * hi];
      for (int kk = 0; kk < 64; kk += 4) {
        v2f a = ldA(ATT, LDT, ti * 16, kk, lh, hi);
        v2f b = ldB(Ut, LDV, kk, tj * 16, lh, hi);
        acc = WMMA_F32(a, b, acc);
      }
      int jl = tj * 16 + lh;
#pragma unroll
      for (int r = 0; r < 8; ++r) {
        int il = ti * 16 + r + 8 * hi;
        out[((size_t)(tbase + il) * HV_N + head) * 128 + split * WV + jl] =
            scaleO * acc[r];
      }
    }
    __syncthreads();
    // ---- S = e^{gcL} S + Ktilde^T @ U,  Ktilde[j] = e^{gcL-gc_j} k_j ----
    {
      float egcl = egc[CH_L - 1];
      int di0 = wid >> 1, di1 = di0 + 4, tjv = wid & 1;
      int jl = tjv * 16 + lh;
      v8f acc0, acc1;
#pragma unroll
      for (int r = 0; r < 8; ++r) {
        acc0[r] = egcl * S[(di0 * 16 + r + 8 * hi) * LDV + jl];
        acc1[r] = egcl * S[(di1 * 16 + r + 8 * hi) * LDV + jl];
      }
      for (int kk = 0; kk < 64; kk += 4) {
        int k = kk + 2 * hi;
        v2f b = ldB(Ut, LDV, kk, tjv * 16, lh, hi);
        v2f a0, a1;  // A = Ktilde^T tile: A[d][l] = escl[l] * Kt[l][d]
        float e0 = escl[k], e1 = escl[k + 1];
        a0.x = e0 * Kt[k * LDQ + di0 * 16 + lh];
        a0.y = e1 * Kt[(k + 1) * LDQ + di0 * 16 + lh];
        a1.x = e0 * Kt[k * LDQ + di1 * 16 + lh];
        a1.y = e1 * Kt[(k + 1) * LDQ + di1 * 16 + lh];
        acc0 = WMMA_F32(a0, b, acc0);
        acc1 = WMMA_F32(a1, b, acc1);
      }
#pragma unroll
      for (int r = 0; r < 8; ++r) {
        S[(di0 * 16 + r + 8 * hi) * LDV + jl] = acc0[r];
        S[(di1 * 16 + r + 8 * hi) * LDV + jl] = acc1[r];
      }
    }
  }
}

extern "C" void kernel_launch(void* const* d_in, const int* in_sizes, int n_in,
                              void* d_out, int out_size, void* d_ws, size_t ws_size,
                              hipStream_t stream) {
  (void)in_sizes; (void)n_in; (void)d_ws; (void)ws_size; (void)out_size;
  const float* mixed = (const float*)d_in[0];
  const float* bg    = (const float*)d_in[1];
  const float* ag    = (const float*)d_in[2];
  const float* cw    = (const float*)d_in[3];
  const float* dtb   = (const float*)d_in[4];
  const float* alog  = (const float*)d_in[5];
  float* out = (float*)d_out;
  gdn_prefill_kernel<<<dim3(HV_N * NSPLIT), dim3(256), 0, stream>>>(
      mixed, bg, ag, cw, dtb, alog, out);
}